// Propagate_54958401519776
// MI455X (gfx1250) — compile-verified
//
#include <hip/hip_runtime.h>

#define FDIM 128   // feature dim (fixed by the reference)

// ---------------- setup kernels ----------------

__global__ void zero_u32_kernel(unsigned* __restrict__ p, int n) {
    int i = blockIdx.x * blockDim.x + threadIdx.x;
    if (i < n) p[i] = 0u;
}

__global__ void deg_count_kernel(const int* __restrict__ col, int E,
                                 unsigned* __restrict__ deg) {
    int e = blockIdx.x * blockDim.x + threadIdx.x;
    if (e < E) atomicAdd(&deg[col[e]], 1u);
}

// Block-local exclusive scan of deg -> excl, per-block totals -> bsum.
__global__ void scan_blocks_kernel(const unsigned* __restrict__ deg,
                                   unsigned* __restrict__ excl,
                                   unsigned* __restrict__ bsum, int n) {
    __shared__ unsigned s[256];
    int i = blockIdx.x * 256 + threadIdx.x;
    unsigned v = (i < n) ? deg[i] : 0u;
    s[threadIdx.x] = v;
    __syncthreads();
    for (int off = 1; off < 256; off <<= 1) {
        unsigned t = (threadIdx.x >= (unsigned)off) ? s[threadIdx.x - off] : 0u;
        __syncthreads();
        s[threadIdx.x] += t;
        __syncthreads();
    }
    if (i < n) excl[i] = s[threadIdx.x] - v;           // exclusive within block
    if (threadIdx.x == 255) bsum[blockIdx.x] = s[255]; // block total
}

// Single-block exclusive scan of the block totals (nb <= 512).
__global__ void scan_bsum_kernel(unsigned* __restrict__ bsum, int nb) {
    __shared__ unsigned s[512];
    unsigned v = ((int)threadIdx.x < nb) ? bsum[threadIdx.x] : 0u;
    s[threadIdx.x] = v;
    __syncthreads();
    for (int off = 1; off < 512; off <<= 1) {
        unsigned t = (threadIdx.x >= (unsigned)off) ? s[threadIdx.x - off] : 0u;
        __syncthreads();
        s[threadIdx.x] += t;
        __syncthreads();
    }
    if ((int)threadIdx.x < nb) bsum[threadIdx.x] = s[threadIdx.x] - v; // exclusive
}

__global__ void finalize_offsets_kernel(unsigned* __restrict__ excl,
                                        const unsigned* __restrict__ bsum,
                                        unsigned* __restrict__ cursor, int n) {
    int i = blockIdx.x * blockDim.x + threadIdx.x;
    if (i < n) {
        unsigned o = excl[i] + bsum[i >> 8];
        excl[i]   = o;   // global exclusive offset
        cursor[i] = o;   // insert cursor for counting sort
    }
}

__global__ void compute_dinv_kernel(const unsigned* __restrict__ deg,
                                    float* __restrict__ dinv, int n) {
    int i = blockIdx.x * blockDim.x + threadIdx.x;
    if (i < n) {
        unsigned d = deg[i];
        dinv[i] = d ? rsqrtf((float)d) : 0.0f;
    }
}

// Counting-sort edges by destination (col); pack (src, weight) into one
// 8-byte record so the hop loop reads each edge with a single B64 load.
__global__ void fill_csc_kernel(const int* __restrict__ row,
                                const int* __restrict__ col, int E,
                                const float* __restrict__ dinv,
                                unsigned* __restrict__ cursor,
                                int2* __restrict__ edges) {
    int e = blockIdx.x * blockDim.x + threadIdx.x;
    if (e < E) {
        int r = row[e];
        int c = col[e];
        unsigned p = atomicAdd(&cursor[c], 1u);
        int2 rec;
        rec.x = r;
        rec.y = __float_as_int(dinv[c] * dinv[r]);
        edges[p] = rec;   // GLOBAL_STORE_B64
    }
}

// ---------------- hop kernel: one wave32 per node ----------------
// Each lane owns 4 of the 128 dims (float4 -> GLOBAL_LOAD/STORE_B128).
// Edge records are read 32-at-a-time per wave (one B64 per lane) and
// broadcast via shfl. Pure gather: no float atomics, each output row
// written exactly once (deg-0 nodes correctly emit zeros).
__global__ __launch_bounds__(256) void hop_kernel(
        const float* __restrict__ hin, float* __restrict__ hout,
        const unsigned* __restrict__ offs, const unsigned* __restrict__ deg,
        const int2* __restrict__ edges, int n) {
    const int lane = threadIdx.x & 31;
    // node is wave-uniform; readfirstlane makes it provably scalar so
    // offs[node]/deg[node] lower to s_load (constant cache) instead of vmem.
    const int node = __builtin_amdgcn_readfirstlane(
        blockIdx.x * 8 + (threadIdx.x >> 5));
    if (node >= n) return;

    const unsigned start = offs[node];
    const unsigned cnt   = deg[node];
    const int dbase = lane * 4;

    float a0 = 0.f, a1 = 0.f, a2 = 0.f, a3 = 0.f;

    for (unsigned base = 0; base < cnt; base += 32) {
        const unsigned idx = base + lane;
        int   s  = 0;
        float ww = 0.f;
        if (idx < cnt) {
            const int2 rec = edges[start + idx];   // GLOBAL_LOAD_B64
            s  = rec.x;
            ww = __int_as_float(rec.y);
            // Pull the rows this wave is about to consume L2 -> L0.
            const float* rp = hin + (size_t)s * FDIM;
            __builtin_prefetch(rp, 0, 3);          // global_prefetch_b8
            __builtin_prefetch(rp + 64, 0, 3);     // second half of 512B row
        }
        const int m = (int)((cnt - base < 32u) ? (cnt - base) : 32u);
        int j = 0;
        // 2x unroll: two b128 gathers in flight before the FMA chain waits.
        for (; j + 1 < m; j += 2) {
            const int   s0 = __shfl(s, j, 32);
            const float w0 = __shfl(ww, j, 32);
            const int   s1 = __shfl(s, j + 1, 32);
            const float w1 = __shfl(ww, j + 1, 32);
            const float4 v0 = *(const float4*)(hin + (size_t)s0 * FDIM + dbase);
            const float4 v1 = *(const float4*)(hin + (size_t)s1 * FDIM + dbase);
            a0 = fmaf(w0, v0.x, a0);
            a1 = fmaf(w0, v0.y, a1);
            a2 = fmaf(w0, v0.z, a2);
            a3 = fmaf(w0, v0.w, a3);
            a0 = fmaf(w1, v1.x, a0);
            a1 = fmaf(w1, v1.y, a1);
            a2 = fmaf(w1, v1.z, a2);
            a3 = fmaf(w1, v1.w, a3);
        }
        if (j < m) {
            const int   sj = __shfl(s, j, 32);
            const float wj = __shfl(ww, j, 32);
            const float4 v = *(const float4*)(hin + (size_t)sj * FDIM + dbase);
            a0 = fmaf(wj, v.x, a0);
            a1 = fmaf(wj, v.y, a1);
            a2 = fmaf(wj, v.z, a2);
            a3 = fmaf(wj, v.w, a3);
        }
    }
    float4 o; o.x = a0; o.y = a1; o.z = a2; o.w = a3;
    *(float4*)(hout + (size_t)node * FDIM + dbase) = o;   // GLOBAL_STORE_B128
}

// ---------------- launcher ----------------

extern "C" void kernel_launch(void* const* d_in, const int* in_sizes, int n_in,
                              void* d_out, int out_size, void* d_ws, size_t ws_size,
                              hipStream_t stream) {
    (void)n_in; (void)out_size; (void)ws_size;

    const float* x  = (const float*)d_in[0];
    const int*   ei = (const int*)d_in[1];   // integer input -> const int* (per contract)
    const int N = in_sizes[0] / FDIM;        // 100000
    const int E = in_sizes[1] / 2;           // 1600000
    const int* row = ei;                     // edge_index[0]
    const int* col = ei + E;                 // edge_index[1]
    float* out = (float*)d_out;

    // Carve the workspace (256B-aligned chunks).
    char* ws = (char*)d_ws;
    auto carve = [&](size_t bytes) -> void* {
        void* p = (void*)ws;
        ws += (bytes + 255) & ~(size_t)255;
        return p;
    };
    const int nblk = (N + 255) / 256;                      // 391 blocks
    unsigned* deg    = (unsigned*)carve((size_t)N * 4);
    unsigned* offs   = (unsigned*)carve((size_t)N * 4);
    unsigned* cursor = (unsigned*)carve((size_t)N * 4);
    unsigned* bsum   = (unsigned*)carve((size_t)nblk * 4);
    float*    dinv   = (float*)carve((size_t)N * 4);
    int2*     edges  = (int2*)carve((size_t)E * 8);
    float*    t1     = (float*)carve((size_t)N * FDIM * 4);

    const int eblk = (E + 255) / 256;

    // 1) degree histogram
    zero_u32_kernel<<<nblk, 256, 0, stream>>>(deg, N);
    deg_count_kernel<<<eblk, 256, 0, stream>>>(col, E, deg);

    // 2) exclusive scan -> CSC offsets + cursors
    scan_blocks_kernel<<<nblk, 256, 0, stream>>>(deg, offs, bsum, N);
    scan_bsum_kernel<<<1, 512, 0, stream>>>(bsum, nblk);
    finalize_offsets_kernel<<<nblk, 256, 0, stream>>>(offs, bsum, cursor, N);

    // 3) normalization weights + counting-sort edges by destination
    compute_dinv_kernel<<<nblk, 256, 0, stream>>>(deg, dinv, N);
    fill_csc_kernel<<<eblk, 256, 0, stream>>>(row, col, E, dinv, cursor, edges);

    // 4) three gather hops, ping-pong x -> out -> t1 -> out (L2-resident)
    const int hblk = (N + 7) / 8;   // 8 waves (nodes) per 256-thread block
    hop_kernel<<<hblk, 256, 0, stream>>>(x,   out, offs, deg, edges, N);
    hop_kernel<<<hblk, 256, 0, stream>>>(out, t1,  offs, deg, edges, N);
    hop_kernel<<<hblk, 256, 0, stream>>>(t1,  out, offs, deg, edges, N);
}